// NeuralFM_38809324487395
// MI455X (gfx1250) — compile-verified
//
#include <hip/hip_runtime.h>

typedef float v2f __attribute__((ext_vector_type(2)));
typedef float v8f __attribute__((ext_vector_type(8)));

#define NFM_BATCH 16384
#define NFM_NF    50
#define NFM_HID   64
#define NFM_L1    128
#define NFM_L2    64
#define NFM_ROWS  16   // batch rows per block (one WMMA M-tile)

__global__ __launch_bounds__(128) void neuralfm_fused_kernel(
    const int*   __restrict__ features,   // [B, 50]
    const float* __restrict__ emb,        // [1M, 64]
    const float* __restrict__ bias_tab,   // [1M, 1]
    const float* __restrict__ w_bias,     // [1,1]
    const float* __restrict__ W1,         // [64,128]
    const float* __restrict__ b1,         // [128]
    const float* __restrict__ W2,         // [128,64]
    const float* __restrict__ b2,         // [64]
    const float* __restrict__ Wp,         // [64,1]
    const float* __restrict__ bp,         // [1]
    float*       __restrict__ out)        // [B,1]
{
    __shared__ float A_s [NFM_ROWS][NFM_HID + 4];   // fm tile, pad->4r+c banks
    __shared__ float H1_s[NFM_ROWS][NFM_L1  + 4];   // relu(fm@W1+b1)
    __shared__ float H2_s[NFM_ROWS][NFM_L2  + 4];   // relu(h1@W2+b2)
    __shared__ float fb_s[NFM_ROWS];                // first-order bias per row

    const int tid  = threadIdx.x;
    const int lane = tid & 31;
    const int wv   = tid >> 5;               // wave id 0..3
    const int b0   = blockIdx.x * NFM_ROWS;  // first batch row of this block
    const int kcol = lane << 1;              // 2 f32 per lane over HID=64

    // ---------------- Phase 1: gather + FM bi-interaction + bias ----------------
    // Wave wv owns rows wv*4..wv*4+3; 4 rows pipelined for memory-level parallelism.
    {
        const int r0 = wv * 4;
        float sx[4], sy[4], qx[4], qy[4], fb[4];
        int idx[4];
#pragma unroll
        for (int r = 0; r < 4; ++r) {
            sx[r] = sy[r] = qx[r] = qy[r] = fb[r] = 0.f;
            idx[r] = features[(b0 + r0 + r) * NFM_NF];
        }
        for (int f = 0; f < NFM_NF; ++f) {
            int nidx[4];
#pragma unroll
            for (int r = 0; r < 4; ++r) {
                nidx[r] = (f + 1 < NFM_NF) ? features[(b0 + r0 + r) * NFM_NF + f + 1]
                                           : idx[r];
                // gfx1250 global_prefetch_b8: pull next field's 256B row toward L2/L0
                __builtin_prefetch(&emb[(long)nidx[r] * NFM_HID + kcol], 0, 3);
            }
#pragma unroll
            for (int r = 0; r < 4; ++r) {
                const float2 e = *(const float2*)&emb[(long)idx[r] * NFM_HID + kcol];
                sx[r] += e.x;        sy[r] += e.y;
                qx[r] += e.x * e.x;  qy[r] += e.y * e.y;
                fb[r] += bias_tab[idx[r]];   // lane-uniform addr -> single request
            }
#pragma unroll
            for (int r = 0; r < 4; ++r) idx[r] = nidx[r];
        }
#pragma unroll
        for (int r = 0; r < 4; ++r) {
            A_s[r0 + r][kcol]     = 0.5f * (sx[r] * sx[r] - qx[r]);
            A_s[r0 + r][kcol + 1] = 0.5f * (sy[r] * sy[r] - qy[r]);
            if (lane == 0) fb_s[r0 + r] = fb[r];
        }
    }
    __syncthreads();

    // Fragment lane decode (ISA 7.12.2, f32 16x16x4):
    //   A: lane L -> row L&15; VGPR0/1 = K + 2*(L>=16) , +1
    //   B: lane L -> col L&15; VGPR0/1 = K + 2*(L>=16) , +1
    //   C/D: VGPR v -> M = v + 8*(L>=16), N = L&15
    const int frow  = lane & 15;
    const int khalf = (lane >> 4) << 1;
    const int mhalf = (lane >> 4) << 3;

    // ---------------- Layer 1: h1 = relu(fm @ W1 + b1), [16,64]@[64,128] ----------------
#pragma unroll
    for (int t = 0; t < 2; ++t) {
        const int n0 = (wv * 2 + t) * 16;    // 8 N-tiles over 4 waves
        v8f c = {};
#pragma unroll 4
        for (int k0 = 0; k0 < NFM_HID; k0 += 4) {
            const float2 af = *(const float2*)&A_s[frow][k0 + khalf];
            v2f a; a.x = af.x; a.y = af.y;
            const int bk = k0 + khalf;
            v2f b;
            b.x = W1[ bk      * NFM_L1 + n0 + frow];
            b.y = W1[(bk + 1) * NFM_L1 + n0 + frow];
            c = __builtin_amdgcn_wmma_f32_16x16x4_f32(false, a, false, b,
                                                      (short)0, c, false, false);
        }
#pragma unroll
        for (int v = 0; v < 8; ++v) {
            const int m = v + mhalf;
            const int n = n0 + frow;
            H1_s[m][n] = fmaxf(c[v] + b1[n], 0.f);
        }
    }
    __syncthreads();

    // ---------------- Layer 2: h2 = relu(h1 @ W2 + b2), [16,128]@[128,64] ----------------
    {
        const int n0 = wv * 16;              // 4 N-tiles over 4 waves
        v8f c = {};
#pragma unroll 4
        for (int k0 = 0; k0 < NFM_L1; k0 += 4) {
            const float2 af = *(const float2*)&H1_s[frow][k0 + khalf];
            v2f a; a.x = af.x; a.y = af.y;
            const int bk = k0 + khalf;
            v2f b;
            b.x = W2[ bk      * NFM_L2 + n0 + frow];
            b.y = W2[(bk + 1) * NFM_L2 + n0 + frow];
            c = __builtin_amdgcn_wmma_f32_16x16x4_f32(false, a, false, b,
                                                      (short)0, c, false, false);
        }
#pragma unroll
        for (int v = 0; v < 8; ++v) {
            const int m = v + mhalf;
            const int n = n0 + frow;
            H2_s[m][n] = fmaxf(c[v] + b2[n], 0.f);
        }
    }
    __syncthreads();

    // ---------------- Final: pred + feature_bias + global_bias ----------------
    if (tid < NFM_ROWS) {
        float acc = 0.f;
#pragma unroll 8
        for (int j = 0; j < NFM_L2; ++j) acc += H2_s[tid][j] * Wp[j];
        out[b0 + tid] = acc + bp[0] + fb_s[tid] + w_bias[0];
    }
}

extern "C" void kernel_launch(void* const* d_in, const int* in_sizes, int n_in,
                              void* d_out, int out_size, void* d_ws, size_t ws_size,
                              hipStream_t stream) {
    const int*   features = (const int*)  d_in[0];
    // d_in[1] = labels (unused by forward output)
    const float* emb      = (const float*)d_in[2];
    const float* bias_tab = (const float*)d_in[3];
    const float* w_bias   = (const float*)d_in[4];
    const float* W1       = (const float*)d_in[5];
    const float* b1       = (const float*)d_in[6];
    const float* W2       = (const float*)d_in[7];
    const float* b2       = (const float*)d_in[8];
    const float* Wp       = (const float*)d_in[9];
    const float* bp       = (const float*)d_in[10];
    float* out = (float*)d_out;

    dim3 grid(NFM_BATCH / NFM_ROWS);   // 1024 blocks
    dim3 block(128);                   // 4 wave32s
    neuralfm_fused_kernel<<<grid, block, 0, stream>>>(
        features, emb, bias_tab, w_bias, W1, b1, W2, b2, Wp, bp, out);
}